// MapEncoder_75453985456550
// MI455X (gfx1250) — compile-verified
//
#include <hip/hip_runtime.h>

// Output: [B=4, 128, 128, 128, C=4] f32.
// C = {occ_a*emb[0][0], occ_a*emb[0][1], occ_b*emb[1][0], occ_b*emb[1][1]}
//
// Pass 1: zero-fill d_out (128 MiB) using CDNA5 async LDS->global b128 stores
//         (GLOBAL_STORE_ASYNC_FROM_LDS_B128, ASYNCcnt path).
// Pass 2: per-point scatter; round-nearest-even voxel index (== jnp.round),
//         aligned 8-byte store of the (emb0, emb1) pair. All racing writers
//         store identical values -> plain stores are deterministic & correct.

typedef int v4i __attribute__((vector_size(16)));                 // int4
typedef __attribute__((address_space(1))) v4i gv4i_t;             // global int4
typedef __attribute__((address_space(3))) v4i lv4i_t;             // LDS int4

__global__ void __launch_bounds__(256) me_zero_async_kernel(float4* __restrict__ out, int n4) {
    __shared__ float4 zbuf[256];

    // zero this thread's 16B LDS staging slot
    zbuf[threadIdx.x] = make_float4(0.f, 0.f, 0.f, 0.f);
    __syncthreads();   // DS writes complete before the async engine reads LDS

    lv4i_t* lds_slot = (lv4i_t*)&zbuf[threadIdx.x];

    int idx    = blockIdx.x * blockDim.x + threadIdx.x;
    int stride = gridDim.x * blockDim.x;
    for (int i = idx; i < n4; i += stride) {
        // async 16B store: this thread's LDS slot -> out[i]; tracked by ASYNCcnt
        __builtin_amdgcn_global_store_async_from_lds_b128(
            (gv4i_t*)(out + i), lds_slot, /*imm offset*/0, /*cpol*/0);
    }
    // drain ASYNCcnt before the wave retires (LDS must stay live until done)
    asm volatile("s_wait_asynccnt 0x0" ::: "memory");
}

__global__ void __launch_bounds__(256) me_scatter_kernel(
        const float* __restrict__ pts,   // [B, N, 3]
        const float* __restrict__ vs,    // [B, 3]
        const float* __restrict__ emb,   // 2 floats for this tensor (uniform -> s_load)
        float* __restrict__ out,         // [B,128,128,128,4]
        int N, int B, int chan_off) {
    int g = blockIdx.x * blockDim.x + threadIdx.x;
    if (g >= B * N) return;
    int b = g / N;

    // streaming read-once point data: non-temporal so it doesn't evict output lines in L2
    const float* p = pts + (size_t)g * 3u;
    float px = __builtin_nontemporal_load(p + 0);
    float py = __builtin_nontemporal_load(p + 1);
    float pz = __builtin_nontemporal_load(p + 2);

    float vx = vs[b * 3 + 0];
    float vy = vs[b * 3 + 1];
    float vz = vs[b * 3 + 2];

    // ijk = round((p - vs*0.5)/vs) - LOW ; LOW = -64 ; rintf == round-half-even == jnp.round
    int i = (int)rintf((px - vx * 0.5f) / vx) + 64;
    int j = (int)rintf((py - vy * 0.5f) / vy) + 64;
    int k = (int)rintf((pz - vz * 0.5f) / vz) + 64;

    if ((unsigned)i < 128u && (unsigned)j < 128u && (unsigned)k < 128u) {
        size_t base = ((((size_t)b * 128u + (unsigned)i) * 128u + (unsigned)j) * 128u
                       + (unsigned)k) * 4u + (unsigned)chan_off;
        float2 v = make_float2(emb[0], emb[1]);
        *(float2*)(out + base) = v;                 // aligned global_store_b64
    }
}

extern "C" void kernel_launch(void* const* d_in, const int* in_sizes, int n_in,
                              void* d_out, int out_size, void* d_ws, size_t ws_size,
                              hipStream_t stream) {
    const float* points_a = (const float*)d_in[0];
    const float* points_b = (const float*)d_in[1];
    const float* vsizes   = (const float*)d_in[2];
    const float* emb      = (const float*)d_in[3];   // [2,2] row-major
    float* out = (float*)d_out;

    const int B = in_sizes[2] / 3;                   // 4
    const int N = in_sizes[0] / (B * 3);             // 200000
    const int n4 = out_size / 4;                     // float4 count

    // 1) zero-fill the dense output via async LDS->global b128 stores
    //    2048 blocks x 256 threads -> each thread issues ~16 async stores
    me_zero_async_kernel<<<2048, 256, 0, stream>>>((float4*)out, n4);

    // 2) scatter each tensor's points; stream order guarantees the fill is done first
    int nthreads = B * N;
    int nblk = (nthreads + 255) / 256;
    me_scatter_kernel<<<nblk, 256, 0, stream>>>(points_a, vsizes, emb + 0, out, N, B, 0);
    me_scatter_kernel<<<nblk, 256, 0, stream>>>(points_b, vsizes, emb + 2, out, N, B, 2);
}